// AttentionWithMemory_31714038514388
// MI455X (gfx1250) — compile-verified
//
#include <hip/hip_runtime.h>
#include <hip/hip_bf16.h>
#include <math.h>

// ---------------------------------------------------------------------------
// AttentionWithMemory for MI455X (gfx1250, wave32, WMMA).
// All heavy GEMMs use V_WMMA_F32_16X16X4_F32 (fp32 matrix path):
//   A 16x4 (v2f/lane), B 4x16 (v2f/lane), C/D 16x16 (v8f/lane).
// Fragment layouts per CDNA5 ISA 7.12.2:
//   A: lane L holds A[L&15][2*(L>>4)+{0,1}]
//   B: lane L holds B[2*(L>>4)+{0,1}][L&15]
//   C: vgpr g of lane L = C[g + 8*(L>>4)][L&15]
// Tile staging uses GLOBAL_LOAD_ASYNC_TO_LDS_B128 (ASYNCcnt) when the
// toolchain exposes the builtin, else plain b128 load + ds_store.
// ---------------------------------------------------------------------------

typedef __attribute__((ext_vector_type(2))) float v2f;
typedef __attribute__((ext_vector_type(8))) float v8f;

__device__ __forceinline__ v8f wmma4(v2f a, v2f b, v8f c) {
  return __builtin_amdgcn_wmma_f32_16x16x4_f32(
      /*neg_a=*/false, a, /*neg_b=*/false, b,
      /*c_mod=*/(short)0, c, /*reuse_a=*/false, /*reuse_b=*/false);
}

#if defined(__HIP_DEVICE_COMPILE__) && defined(__gfx1250__) &&        \
    __has_builtin(__builtin_amdgcn_global_load_async_to_lds_b128) &&  \
    __has_builtin(__builtin_amdgcn_s_wait_asynccnt)
#define USE_ASYNC_LDS 1
#else
#define USE_ASYNC_LDS 0
#endif

typedef int v4i_b128 __attribute__((vector_size(16)));

__device__ __forceinline__ void copy16_g2l(const float* __restrict__ g,
                                           float* l) {
#if USE_ASYNC_LDS
  __builtin_amdgcn_global_load_async_to_lds_b128(
      (__attribute__((address_space(1))) v4i_b128*)g,
      (__attribute__((address_space(3))) v4i_b128*)l, /*offset=*/0,
      /*cpol=*/0);
#else
  *(float4*)l = *(const float4*)g;
#endif
}

__device__ __forceinline__ void async_fence() {
#if USE_ASYNC_LDS
  __builtin_amdgcn_s_wait_asynccnt(0);
#endif
}

#define TK 32  // K-chunk staged in LDS per GEMM step (pad +4 keeps 16B rows)

// ---------------------------------------------------------------------------
// inv L2 row norms: out[row] = 1 / max(||X[row,:]||_2, 1e-12)
// ---------------------------------------------------------------------------
__global__ __launch_bounds__(128) void rownorm_inv_kernel(
    const float* __restrict__ X, float* __restrict__ out, int E) {
  __shared__ float red[128];
  const int row = blockIdx.x;
  const float* x = X + (size_t)row * E;
  float s = 0.f;
  for (int i = threadIdx.x; i < E; i += 128) {
    float v = x[i];
    s += v * v;
  }
  red[threadIdx.x] = s;
  __syncthreads();
  for (int off = 64; off > 0; off >>= 1) {
    if (threadIdx.x < off) red[threadIdx.x] += red[threadIdx.x + off];
    __syncthreads();
  }
  if (threadIdx.x == 0) out[row] = 1.0f / fmaxf(sqrtf(red[0]), 1e-12f);
}

// ---------------------------------------------------------------------------
// C[M,N] = A[M,K] @ Bw[N,K]^T + bias[N]      (NT GEMM, fp32 WMMA)
// Block: 128 threads (4 waves, 2x2). WG tile 64x64; wave tile 32x32
// (2x2 fragments -> 4 LDS frag reads per 4 WMMAs).
// ---------------------------------------------------------------------------
__global__ __launch_bounds__(128) void gemm_nt_bias_kernel(
    const float* __restrict__ A, const float* __restrict__ Bw,
    const float* __restrict__ bias, float* __restrict__ C,
    int M, int N, int K) {
  __shared__ float As[64][TK + 4];
  __shared__ float Bs[64][TK + 4];

  const int tid = threadIdx.x;
  const int lane = tid & 31;
  const int wave = tid >> 5;
  const int l15 = lane & 15;
  const int hi = lane >> 4;
  const int wr = (wave >> 1) << 5;  // 0 or 32
  const int wc = (wave & 1) << 5;   // 0 or 32
  const int row0 = blockIdx.y * 64;
  const int col0 = blockIdx.x * 64;

  v8f acc00 = {}, acc01 = {}, acc10 = {}, acc11 = {};

  for (int k0 = 0; k0 < K; k0 += TK) {
    for (int i = tid; i < 64 * (TK / 4); i += 128) {
      const int r = i >> 3, kc = (i & 7) << 2;
      copy16_g2l(A + (size_t)(row0 + r) * K + k0 + kc, &As[r][kc]);
      copy16_g2l(Bw + (size_t)(col0 + r) * K + k0 + kc, &Bs[r][kc]);
    }
    if (k0 + TK < K)  // gfx1250 global_prefetch_b8 on next slab
      __builtin_prefetch(A + (size_t)(row0 + (tid & 63)) * K + k0 + TK, 0, 0);
    async_fence();
    __syncthreads();

#pragma unroll
    for (int kk = 0; kk < TK; kk += 4) {
      const int kk2 = kk + (hi << 1);
      v2f a0 = {As[wr + l15][kk2],      As[wr + l15][kk2 + 1]};
      v2f a1 = {As[wr + 16 + l15][kk2], As[wr + 16 + l15][kk2 + 1]};
      v2f b0 = {Bs[wc + l15][kk2],      Bs[wc + l15][kk2 + 1]};
      v2f b1 = {Bs[wc + 16 + l15][kk2], Bs[wc + 16 + l15][kk2 + 1]};
      acc00 = wmma4(a0, b0, acc00);
      acc01 = wmma4(a0, b1, acc01);
      acc10 = wmma4(a1, b0, acc10);
      acc11 = wmma4(a1, b1, acc11);
    }
    __syncthreads();
  }

  const int ccol = col0 + wc + l15;
  const float bi0 = bias[ccol];
  const float bi1 = bias[ccol + 16];
#pragma unroll
  for (int g = 0; g < 8; ++g) {
    const int ra = row0 + wr + g + (hi << 3);
    const int rb = ra + 16;
    C[(size_t)ra * N + ccol]      = acc00[g] + bi0;
    C[(size_t)ra * N + ccol + 16] = acc01[g] + bi1;
    C[(size_t)rb * N + ccol]      = acc10[g] + bi0;
    C[(size_t)rb * N + ccol + 16] = acc11[g] + bi1;
  }
}

// ---------------------------------------------------------------------------
// Fused cosine-similarity GEMM + top-8 selection.
// One WG = 16 query rows; whole 16xE query tile cached in LDS.
// Streams keys in 128-wide tiles; wave owns 32 sim columns (2 accums).
// Per-thread top-8 in registers (8 threads/row, 16 cols each), merged at end.
// ---------------------------------------------------------------------------
#define MT 128
__global__ __launch_bounds__(128) void topk_kernel(
    const float* __restrict__ X, const float* __restrict__ MK,
    const float* __restrict__ hinv, const float* __restrict__ kinv,
    float* __restrict__ tvals, int* __restrict__ tidx, int E, int Mrows) {
  __shared__ float Qs[16][1024];
  __shared__ float Ks[MT][TK + 4];
  __shared__ float Ssim[16][MT];
  __shared__ float cval[16][64];
  __shared__ int   cidx[16][64];
  __shared__ float qiv[16];

  const int tid = threadIdx.x;
  const int lane = tid & 31;
  const int wave = tid >> 5;
  const int l15 = lane & 15;
  const int hi = lane >> 4;
  const int wc = wave << 5;       // wave's sim-column base within MT
  const int r0 = blockIdx.x * 16; // base query row in [0, B*S)

  for (int i = tid; i < 16 * (1024 / 4); i += 128) {
    const int r = i >> 8, c = (i & 255) << 2;
    copy16_g2l(X + (size_t)(r0 + r) * E + c, &Qs[r][c]);
  }
  if (tid < 16) qiv[tid] = hinv[r0 + tid];

  float bv8[8];
  int   bi8[8];
#pragma unroll
  for (int q = 0; q < 8; ++q) { bv8[q] = -1e30f; bi8[q] = 0; }
  async_fence();
  __syncthreads();

  for (int m0 = 0; m0 < Mrows; m0 += MT) {
    v8f acc0 = {}, acc1 = {};
    for (int kt = 0; kt < E; kt += TK) {
      for (int i = tid; i < MT * (TK / 4); i += 128) {
        const int r = i >> 3, kc = (i & 7) << 2;
        copy16_g2l(MK + (size_t)(m0 + r) * E + kt + kc, &Ks[r][kc]);
      }
      async_fence();
      __syncthreads();
#pragma unroll
      for (int kk = 0; kk < TK; kk += 4) {
        const int kk2 = kk + (hi << 1);
        v2f a  = {Qs[l15][kt + kk2], Qs[l15][kt + kk2 + 1]};
        v2f b0 = {Ks[wc + l15][kk2],      Ks[wc + l15][kk2 + 1]};
        v2f b1 = {Ks[wc + 16 + l15][kk2], Ks[wc + 16 + l15][kk2 + 1]};
        acc0 = wmma4(a, b0, acc0);
        acc1 = wmma4(a, b1, acc1);
      }
      __syncthreads();
    }
    if (m0 + MT < Mrows)
      __builtin_prefetch(MK + (size_t)(m0 + MT + tid) * E, 0, 0);

    // scale by both inverse norms -> cosine sims, dump tile
    {
      const int col = wc + l15;
      const float kv0 = kinv[m0 + col];
      const float kv1 = kinv[m0 + col + 16];
#pragma unroll
      for (int g = 0; g < 8; ++g) {
        const int r = g + (hi << 3);
        Ssim[r][col]      = acc0[g] * qiv[r] * kv0;
        Ssim[r][col + 16] = acc1[g] * qiv[r] * kv1;
      }
    }
    __syncthreads();

    // per-thread top-8 maintenance: 8 threads / row, 16 cols each
    {
      const int row = tid & 15;
      const int grp = tid >> 4;
#pragma unroll
      for (int cc = 0; cc < 16; ++cc) {
        const int c = (grp << 4) + cc;
        float v = Ssim[row][c];
        int idx = m0 + c;
#pragma unroll
        for (int q = 0; q < 8; ++q) {
          if (v > bv8[q]) {
            const float tv0 = bv8[q]; const int ti0 = bi8[q];
            bv8[q] = v; bi8[q] = idx;
            v = tv0; idx = ti0;
          }
        }
      }
    }
    __syncthreads();
  }

  // merge 8 partial top-8 lists per row
  {
    const int row = tid & 15;
    const int grp = tid >> 4;
#pragma unroll
    for (int q = 0; q < 8; ++q) {
      cval[row][(grp << 3) + q] = bv8[q];
      cidx[row][(grp << 3) + q] = bi8[q];
    }
  }
  __syncthreads();
  if (tid < 16) {
    const int row = tid;
    for (int j = 0; j < 8; ++j) {
      float best = -1e30f; int bidx = 0; int bpos = 0;
      for (int c = 0; c < 64; ++c) {
        const float v = cval[row][c];
        if (v > best) { best = v; bidx = cidx[row][c]; bpos = c; }
      }
      cval[row][bpos] = -1e31f;
      tvals[(size_t)(r0 + row) * 8 + j] = best;
      tidx [(size_t)(r0 + row) * 8 + j] = bidx;
    }
  }
}

// ---------------------------------------------------------------------------
// Flash attention + memory columns.
// One WG = (b, h, 16 query rows). 4 waves. Keys/values streamed 128 at a
// time; wave owns 32 score cols (2 accums) and 16 hd cols for P*V.
// Online softmax over S, then the 8 memory columns (raw cosine, unscaled)
// folded into the same running max/sum with gathered memory_values.
// ---------------------------------------------------------------------------
#define ST 128
__global__ __launch_bounds__(128) void attn_kernel(
    const float* __restrict__ Q, const float* __restrict__ Kt,
    const float* __restrict__ V, const float* __restrict__ MV,
    const float* __restrict__ tvals, const int* __restrict__ tidx,
    float* __restrict__ ctx, int B, int S, int E, int H) {
  const int hd = 64;
  __shared__ float Qs[16][68];
  __shared__ float Ks[ST][68];
  __shared__ float Vs[ST][68];
  __shared__ float Pt[16][ST];
  __shared__ float mrow[16], lrow[16], rf[16];
  __shared__ float pm[16][8];
  __shared__ int   im[16][8];

  const int tid = threadIdx.x;
  const int lane = tid & 31;
  const int wave = tid >> 5;
  const int l15 = lane & 15;
  const int hi = lane >> 4;
  const int kb = wave << 5;            // score-column base within ST
  const int hcol = (wave << 4) + l15;  // hd column owned for P*V / output

  const int nqt = S >> 4;
  const int qt = blockIdx.x % nqt;
  const int h = (blockIdx.x / nqt) % H;
  const int b = blockIdx.x / (nqt * H);
  const size_t bse = (size_t)b * S * E;
  const int hoff = h * hd;
  const float scale = 0.125f;  // 1/sqrt(64)

  for (int i = tid; i < 16 * (hd / 4); i += 128) {
    const int r = i >> 4, c = (i & 15) << 2;
    copy16_g2l(Q + bse + (size_t)((qt << 4) + r) * E + hoff + c, &Qs[r][c]);
  }
  if (tid < 16) { mrow[tid] = -1e30f; lrow[tid] = 0.f; }
  v8f acc = {};
  async_fence();
  __syncthreads();

  for (int s0 = 0; s0 < S; s0 += ST) {
    for (int i = tid; i < ST * (hd / 4); i += 128) {
      const int r = i >> 4, c = (i & 15) << 2;
      copy16_g2l(Kt + bse + (size_t)(s0 + r) * E + hoff + c, &Ks[r][c]);
      copy16_g2l(V  + bse + (size_t)(s0 + r) * E + hoff + c, &Vs[r][c]);
    }
    async_fence();
    __syncthreads();

    // scores: Q[16x64] x K^T -> wave's 16x32 tile
    v8f sc0 = {}, sc1 = {};
#pragma unroll
    for (int kk = 0; kk < 64; kk += 4) {
      const int kk2 = kk + (hi << 1);
      v2f a  = {Qs[l15][kk2], Qs[l15][kk2 + 1]};
      v2f b0 = {Ks[kb + l15][kk2],      Ks[kb + l15][kk2 + 1]};
      v2f b1 = {Ks[kb + 16 + l15][kk2], Ks[kb + 16 + l15][kk2 + 1]};
      sc0 = wmma4(a, b0, sc0);
      sc1 = wmma4(a, b1, sc1);
    }
    {
      const int col = kb + l15;
#pragma unroll
      for (int g = 0; g < 8; ++g) {
        Pt[g + (hi << 3)][col]      = sc0[g] * scale;
        Pt[g + (hi << 3)][col + 16] = sc1[g] * scale;
      }
    }
    __syncthreads();

    // online softmax row update
    if (tid < 16) {
      const int row = tid;
      float mx = mrow[row];
      for (int c = 0; c < ST; ++c) mx = fmaxf(mx, Pt[row][c]);
      const float f = __expf(mrow[row] - mx);
      float l = lrow[row] * f;
      for (int c = 0; c < ST; ++c) {
        const float p = __expf(Pt[row][c] - mx);
        Pt[row][c] = p;
        l += p;
      }
      mrow[row] = mx; lrow[row] = l; rf[row] = f;
    }
    __syncthreads();

#pragma unroll
    for (int g = 0; g < 8; ++g) acc[g] *= rf[g + (hi << 3)];

    // acc += P[16xST] x V[STx16(wave)]
#pragma unroll
    for (int kk = 0; kk < ST; kk += 4) {
      const int kk2 = kk + (hi << 1);
      v2f a  = {Pt[l15][kk2], Pt[l15][kk2 + 1]};
      v2f bb = {Vs[kk2][hcol], Vs[kk2 + 1][hcol]};
      acc = wmma4(a, bb, acc);
    }
    __syncthreads();
  }

  // fold in the 8 memory columns (raw cosine sims, NOT scaled)
  if (tid < 16) {
    const int row = tid;
    const size_t bs = (size_t)b * S + (qt << 4) + row;
    float mx = mrow[row];
    float v8[8]; int i8[8];
#pragma unroll
    for (int j = 0; j < 8; ++j) {
      v8[j] = tvals[bs * 8 + j];
      i8[j] = tidx[bs * 8 + j];
      mx = fmaxf(mx, v8[j]);
    }
    const float f = __expf(mrow[row] - mx);
    float l = lrow[row] * f;
#pragma unroll
    for (int j = 0; j < 8; ++j) {
      const float e = __expf(v8[j] - mx);
      pm[row][j] = e;
      im[row][j] = i8[j];
      l += e;
    }
    mrow[row] = mx; lrow[row] = l; rf[row] = f;
  }
  __syncthreads();

#pragma unroll
  for (int g = 0; g < 8; ++g) acc[g] *= rf[g + (hi << 3)];

  {
#pragma unroll
    for (int g = 0; g < 8; ++g) {
      const int r = g + (hi << 3);
      float sum = 0.f;
#pragma unroll
      for (int j = 0; j < 8; ++j)
        sum += pm[r][j] * MV[(size_t)im[r][j] * E + hoff + hcol];
      acc[g] += sum;
    }
#pragma unroll
    for (int g = 0; g < 8; ++g) {
      const int r = g + (hi << 3);
      ctx[bse + (size_t)((qt << 4) + r) * E + hoff + hcol] = acc[g] / lrow[r];
    }
  }
}

// ---------------------------------------------------------------------------
extern "C" void kernel_launch(void* const* d_in, const int* in_sizes, int n_in,
                              void* d_out, int out_size, void* d_ws,
                              size_t ws_size, hipStream_t stream) {
  const float* hidden = (const float*)d_in[0];
  const float* mkeys  = (const float*)d_in[1];
  const float* mvals  = (const float*)d_in[2];
  const float* Wq = (const float*)d_in[3];
  const float* bq = (const float*)d_in[4];
  const float* Wk = (const float*)d_in[5];
  const float* bk = (const float*)d_in[6];
  const float* Wv = (const float*)d_in[7];
  const float* bv = (const float*)d_in[8];
  const float* Wo = (const float*)d_in[9];
  const float* bo = (const float*)d_in[10];
  // d_in[11] = top_k (== 8, compile-time specialized)

  const int B = 2, S = 2048, E = 1024, H = 16, M = 16384;
  const int BS = B * S;

  float* ws = (float*)d_ws;
  float* Qp   = ws;                         // BS*E
  float* Kp   = Qp + (size_t)BS * E;        // BS*E
  float* Vp   = Kp + (size_t)BS * E;        // BS*E
  float* Ct   = Vp + (size_t)BS * E;        // BS*E (ctx)
  float* kinv = Ct + (size_t)BS * E;        // M
  float* hinv = kinv + M;                   // BS
  float* tvp  = hinv + BS;                  // BS*8
  int*   tip  = (int*)(tvp + (size_t)BS * 8);  // BS*8

  rownorm_inv_kernel<<<M, 128, 0, stream>>>(mkeys, kinv, E);
  rownorm_inv_kernel<<<BS, 128, 0, stream>>>(hidden, hinv, E);

  dim3 gg(E / 64, BS / 64);
  gemm_nt_bias_kernel<<<gg, 128, 0, stream>>>(hidden, Wq, bq, Qp, BS, E, E);
  gemm_nt_bias_kernel<<<gg, 128, 0, stream>>>(hidden, Wk, bk, Kp, BS, E, E);
  gemm_nt_bias_kernel<<<gg, 128, 0, stream>>>(hidden, Wv, bv, Vp, BS, E, E);

  topk_kernel<<<BS / 16, 128, 0, stream>>>(hidden, mkeys, hinv, kinv, tvp,
                                           tip, E, M);

  attn_kernel<<<B * H * (S / 16), 128, 0, stream>>>(Qp, Kp, Vp, mvals, tvp,
                                                    tip, Ct, B, S, E, H);

  gemm_nt_bias_kernel<<<gg, 128, 0, stream>>>(Ct, Wo, bo, (float*)d_out, BS,
                                              E, E);
}